// Attention_86182813762351
// MI455X (gfx1250) — compile-verified
//
#include <hip/hip_runtime.h>
#include <hip/hip_bf16.h>
#include <math.h>

#define B_ 16
#define S_ 4096
#define H_ 1024
#define SCHUNK 128
#define NCHUNK (S_ / SCHUNK)   // 32

typedef __attribute__((ext_vector_type(2))) float v2f;
typedef __attribute__((ext_vector_type(8))) float v8f;

// ---------------------------------------------------------------------------
// K0: dec_fea = s_t_hat @ W_dec^T + b_dec   via V_WMMA_F32_16X16X4_F32
// One wave per 16-wide N tile (64 waves total). M=16 == B exactly.
// A (16x4 f32) layout: lanes 0-15 hold M=lane,K={k,k+1}; lanes 16-31 M=lane-16,K={k+2,k+3}
// B (4x16 f32) layout: mirrored — lane half selects K pair, lane%16 selects N.
// ---------------------------------------------------------------------------
__global__ __launch_bounds__(32) void dec_proj_wmma_kernel(
    const float* __restrict__ s_t_hat,   // [16,1024]
    const float* __restrict__ W_dec,     // [1024,1024]
    const float* __restrict__ b_dec,     // [1024]
    float* __restrict__ dec_fea)         // [16,1024] (workspace)
{
    const int lane = threadIdx.x;        // 0..31
    const int half = lane >> 4;          // 0 -> K {k,k+1}; 1 -> K {k+2,k+3}
    const int l16  = lane & 15;
    const int n0   = blockIdx.x << 4;    // 64 blocks cover N=1024

    v8f c = {};
    const float* ap_base = s_t_hat + l16 * H_;            // row M = l16
    const float* bp_base = W_dec + (size_t)(n0 + l16) * H_; // W_dec row = output col N

    for (int k = 0; k < H_; k += 4) {
        const int ka = k + half * 2;
        v2f a, b;
        a.x = ap_base[ka];     a.y = ap_base[ka + 1];
        b.x = bp_base[ka];     b.y = bp_base[ka + 1];
        // D = A(16x4) x B(4x16) + C
        c = __builtin_amdgcn_wmma_f32_16x16x4_f32(
                /*neg_a=*/false, a, /*neg_b=*/false, b,
                /*c_mod=*/(short)0, c, /*reuse_a=*/false, /*reuse_b=*/false);
    }

    // C/D layout: VGPR r holds M = r (lanes 0-15) or M = r+8 (lanes 16-31), N = n0 + l16
    const int n = n0 + l16;
    const float bias = b_dec[n];
#pragma unroll
    for (int r = 0; r < 8; ++r) {
        const int m = r + half * 8;
        dec_fea[m * H_ + n] = c[r] + bias;
    }
}

// ---------------------------------------------------------------------------
// K1: scores[b,s] = sum_h v[h] * tanh(ef[b,s,h] + dec_fea[b,h] + cov[b,s]*W_c[h])
// One wave per row; lane loads float4; coalesced 512B per wave-load.
// Streams encoder_feature (268 MB) -> bandwidth-bound.
// ---------------------------------------------------------------------------
__global__ __launch_bounds__(256) void scores_kernel(
    const float* __restrict__ ef,        // [B*S, H]
    const float* __restrict__ dec_fea,   // [B, H]
    const float* __restrict__ coverage,  // [B, S]
    const float* __restrict__ W_c,       // [H]
    const float* __restrict__ v,         // [H]
    float* __restrict__ scores)          // [B, S]
{
    const int wave = threadIdx.x >> 5;
    const int lane = threadIdx.x & 31;
    const long r = (long)blockIdx.x * 8 + wave;      // row in [0, B*S)
    const int  b = (int)(r >> 12);                   // r / S_

    const float cov = coverage[r];
    const float* __restrict__ efr = ef + r * H_;
    const float* __restrict__ df  = dec_fea + (size_t)b * H_;

    float sum = 0.0f;
#pragma unroll
    for (int c = 0; c < 8; ++c) {
        const int h = c * 128 + lane * 4;
        const float4 e4 = *(const float4*)(efr + h);
        const float4 d4 = *(const float4*)(df + h);
        const float4 v4 = *(const float4*)(v + h);
        const float4 w4 = *(const float4*)(W_c + h);
        sum += v4.x * tanhf(e4.x + d4.x + cov * w4.x);
        sum += v4.y * tanhf(e4.y + d4.y + cov * w4.y);
        sum += v4.z * tanhf(e4.z + d4.z + cov * w4.z);
        sum += v4.w * tanhf(e4.w + d4.w + cov * w4.w);
    }
    // wave32 butterfly reduce (deterministic)
#pragma unroll
    for (int off = 16; off > 0; off >>= 1)
        sum += __shfl_xor(sum, off, 32);
    if (lane == 0) scores[r] = sum;
}

// ---------------------------------------------------------------------------
// K2: masked + renormalized softmax over S per batch row. One block per b.
// ---------------------------------------------------------------------------
__global__ __launch_bounds__(256) void softmax_kernel(
    const float* __restrict__ scores,    // [B, S]
    const float* __restrict__ mask,      // [B, S]
    float* __restrict__ attn1,
    float* __restrict__ attn2)
{
    __shared__ float red[256];
    const int b = blockIdx.x;
    const int tid = threadIdx.x;
    const float* __restrict__ sr = scores + (size_t)b * S_;
    const float* __restrict__ mr = mask   + (size_t)b * S_;

    float loc[16];
    float m = -3.402823466e38f;
#pragma unroll
    for (int i = 0; i < 16; ++i) {
        loc[i] = sr[tid + i * 256];
        m = fmaxf(m, loc[i]);
    }
    red[tid] = m; __syncthreads();
    for (int s = 128; s > 0; s >>= 1) {
        if (tid < s) red[tid] = fmaxf(red[tid], red[tid + s]);
        __syncthreads();
    }
    m = red[0]; __syncthreads();

    float se = 0.0f, sem = 0.0f;
#pragma unroll
    for (int i = 0; i < 16; ++i) {
        const float e = expf(loc[i] - m);
        loc[i] = e;
        se  += e;
        sem += e * mr[tid + i * 256];
    }
    red[tid] = se; __syncthreads();
    for (int s = 128; s > 0; s >>= 1) {
        if (tid < s) red[tid] += red[tid + s];
        __syncthreads();
    }
    se = red[0]; __syncthreads();

    red[tid] = sem; __syncthreads();
    for (int s = 128; s > 0; s >>= 1) {
        if (tid < s) red[tid] += red[tid + s];
        __syncthreads();
    }
    sem = red[0];

    const float inv   = 1.0f / se;                 // softmax denom
    const float dinv  = 1.0f / (sem * inv + 1e-20f); // renorm denom
#pragma unroll
    for (int i = 0; i < 16; ++i) {
        const int s = tid + i * 256;
        const float a = loc[i] * inv * mr[s] * dinv;
        attn1[(size_t)b * S_ + s] = a;
        attn2[(size_t)b * S_ + s] = a;
    }
}

// ---------------------------------------------------------------------------
// K3: partial c_t over S-chunks. Streams encoder_outputs (268 MB).
// Block = (s-chunk, b); thread owns float4 of H; attn chunk staged in LDS.
// ---------------------------------------------------------------------------
__global__ __launch_bounds__(256) void ct_partial_kernel(
    const float* __restrict__ enc_out,   // [B, S, H]
    const float* __restrict__ attn,      // [B, S]
    float* __restrict__ partial)         // [B, NCHUNK, H] (workspace)
{
    __shared__ float a_s[SCHUNK];
    const int chunk = blockIdx.x;        // 0..NCHUNK-1
    const int b     = blockIdx.y;        // 0..B-1
    const int tid   = threadIdx.x;
    const int s0    = chunk * SCHUNK;

    if (tid < SCHUNK) a_s[tid] = attn[(size_t)b * S_ + s0 + tid];
    __syncthreads();

    const float* __restrict__ ep =
        enc_out + ((long)b * S_ + s0) * H_ + tid * 4;

    float4 acc = make_float4(0.f, 0.f, 0.f, 0.f);
    for (int s = 0; s < SCHUNK; ++s) {
        const float a = a_s[s];
        const float4 e = *(const float4*)(ep + (long)s * H_);
        acc.x += a * e.x; acc.y += a * e.y;
        acc.z += a * e.z; acc.w += a * e.w;
    }
    *(float4*)(partial + ((long)(b * NCHUNK + chunk)) * H_ + tid * 4) = acc;
}

// ---------------------------------------------------------------------------
// K4: reduce partials -> c_t (fixed order, deterministic)
// ---------------------------------------------------------------------------
__global__ __launch_bounds__(256) void ct_reduce_kernel(
    const float* __restrict__ partial,   // [B, NCHUNK, H]
    float* __restrict__ ct)              // [B, H]
{
    const int idx = blockIdx.x * 256 + threadIdx.x;   // 0..B*H-1
    const int b = idx >> 10;
    const int h = idx & (H_ - 1);
    float s = 0.0f;
#pragma unroll
    for (int c = 0; c < NCHUNK; ++c)
        s += partial[((long)(b * NCHUNK + c)) * H_ + h];
    ct[idx] = s;
}

// ---------------------------------------------------------------------------
extern "C" void kernel_launch(void* const* d_in, const int* in_sizes, int n_in,
                              void* d_out, int out_size, void* d_ws, size_t ws_size,
                              hipStream_t stream) {
    const float* s_t_hat  = (const float*)d_in[0];   // [16,1024]
    const float* enc_out  = (const float*)d_in[1];   // [16,4096,1024]
    const float* enc_feat = (const float*)d_in[2];   // [65536,1024]
    const float* mask     = (const float*)d_in[3];   // [16,4096]
    const float* coverage = (const float*)d_in[4];   // [16,4096]
    const float* W_dec    = (const float*)d_in[5];   // [1024,1024]
    const float* b_dec    = (const float*)d_in[6];   // [1024]
    const float* W_c      = (const float*)d_in[7];   // [1024]
    const float* v        = (const float*)d_in[8];   // [1024]

    float* out = (float*)d_out;
    float* out_ct     = out;                               // 16*1024
    float* out_attn1  = out + B_ * H_;                     // 16*4096
    float* out_attn2  = out_attn1 + B_ * S_;               // 16*4096
    float* out_scores = out_attn2 + B_ * S_;               // 16*4096

    float* ws_dec_fea = (float*)d_ws;                      // 16*1024 f32 = 64 KB
    float* ws_partial = ws_dec_fea + B_ * H_;              // 16*32*1024 f32 = 2 MB

    // K0: dec_fea GEMM via WMMA f32 16x16x4 (64 waves)
    dec_proj_wmma_kernel<<<64, 32, 0, stream>>>(s_t_hat, W_dec, b_dec, ws_dec_fea);

    // K1: fused add+tanh+dot -> raw scores (streams encoder_feature)
    scores_kernel<<<(B_ * S_) / 8, 256, 0, stream>>>(
        enc_feat, ws_dec_fea, coverage, W_c, v, out_scores);

    // K2: masked renormalized softmax
    softmax_kernel<<<B_, 256, 0, stream>>>(out_scores, mask, out_attn1, out_attn2);

    // K3: weighted sum partials (streams encoder_outputs)
    ct_partial_kernel<<<dim3(NCHUNK, B_), 256, 0, stream>>>(
        enc_out, out_attn1, ws_partial);

    // K4: final c_t
    ct_reduce_kernel<<<(B_ * H_) / 256, 256, 0, stream>>>(ws_partial, out_ct);
}